// RLSTBits_27659589386256
// MI455X (gfx1250) — compile-verified
//
#include <hip/hip_runtime.h>
#include <math.h>

// ---------------------------------------------------------------------------
// RLSTBits forward on MI455X (gfx1250, wave32).
// fp32 WMMA (V_WMMA_F32_16X16X4_F32) for all GEMMs; B=32 = 2 M-tiles of 16.
// One full net_step per scan iteration (second net_step of step t == first
// net_step of step t+1, since carry/h are identical). 95 steps, stream-ordered.
// ---------------------------------------------------------------------------

#define kS 48
#define kT 48
#define kB 32
#define kEmb 256
#define kHid 512
#define kInDim 516           /* 2*EMB + 4 */
#define kG3 1536             /* 3*HID     */
#define kVT 16000
#define kNOut 16003
#define kSteps 95            /* S+T-1 */

#define SRC_EOSc 2
#define SRC_NULLc 3
#define TRG_EOSc 2
#define TRG_NULLc 3
#define TRG_PADc 1
#define DISCOUNTc 0.9f
#define Mc 7.0f

// integer state offsets (in ints) inside the int-state block of d_ws
#define IS_WNEW 0
#define IS_I 32
#define IS_J 64
#define IS_TERM 96
#define IS_WRITING 128
#define IS_NAUGHTY 160
#define IS_WTOK 192
#define IS_OUTP 224
#define IS_OLDJ 256
#define IS_READING 288
#define IS_JUSTTERM 320
#define IS_ACNT 352
#define IS_TOTAL 356

typedef float v2f __attribute__((ext_vector_type(2)));
typedef float v8f __attribute__((ext_vector_type(8)));

#define WMMA_F32X4(A, B, C) \
  __builtin_amdgcn_wmma_f32_16x16x4_f32(false, (A), false, (B), (short)0, (C), false, false)

// ---------------------------------------------------------------------------
// Y(32 x N) = X(32 x K) @ W(N x K)^T + bias     (fp32 WMMA 16x16x4)
// One wave (32 threads) per 16-column tile; two M-tiles (rows 0-15 / 16-31).
// A frag (16x4):  lanes 0-15 hold K=k,k+1;  lanes 16-31 hold K=k+2,k+3 (M=lane&15)
// B frag (4x16):  VGPR0/1 = K rows {k,k+1} (lanes 0-15) / {k+2,k+3} (lanes 16-31)
// C/D (16x16):    VGPR v -> row v + 8*(lane>>4), col = lane&15
//
// Out-of-range columns (last tile of the 16003-wide GEMM) clamp the W row
// pointer to row N-1: WMMA output column n depends only on B column n, so the
// garbage stays in never-stored columns. No EXEC divergence in the K-loop.
// Software-pipelined: next slice's loads issue before current slice's WMMAs,
// so waits are partial (loadcnt<=3) and latency overlaps WMMA issue.
// ---------------------------------------------------------------------------
__global__ __launch_bounds__(32)
void gemm32_wmma(const float* __restrict__ X, const float* __restrict__ W,
                 const float* __restrict__ bias, float* __restrict__ Y,
                 int N, int K)
{
  const int lane = threadIdx.x;
  const int l    = lane & 15;
  const int hi   = lane >> 4;
  const int col  = blockIdx.x * 16 + l;
  const int colc = (col < N) ? col : (N - 1);        // clamp: always valid row
  const float* wrow = W + (size_t)colc * (size_t)K;
  const float* x0   = X + (size_t)l * K;
  const float* x1   = X + (size_t)(l + 16) * K;
  const int base = 2 * hi;

  v8f acc0 = {};   // rows 0..15
  v8f acc1 = {};   // rows 16..31

  // prologue: fragments for k = 0
  v2f a0 = *(const v2f*)(x0 + base);
  v2f a1 = *(const v2f*)(x1 + base);
  v2f b  = *(const v2f*)(wrow + base);

#pragma unroll 4
  for (int k = 4; k < K; k += 4) {
    const int kk = k + base;
    v2f na0 = *(const v2f*)(x0 + kk);     // loads for slice k issued first
    v2f na1 = *(const v2f*)(x1 + kk);
    v2f nb  = *(const v2f*)(wrow + kk);
    acc0 = WMMA_F32X4(a0, b, acc0);       // consume slice k-4 while k in flight
    acc1 = WMMA_F32X4(a1, b, acc1);
    a0 = na0; a1 = na1; b = nb;
  }
  acc0 = WMMA_F32X4(a0, b, acc0);         // epilogue: last slice
  acc1 = WMMA_F32X4(a1, b, acc1);

  if (col < N) {
    const float bv = bias[col];
#pragma unroll
    for (int v = 0; v < 8; ++v) {
      const int m = v + 8 * hi;
      Y[(size_t)m        * N + col] = acc0[v] + bv;
      Y[(size_t)(m + 16) * N + col] = acc1[v] + bv;
    }
  }
}

// h[b][c] = (1-z)*n + z*h ;  r,z from sigmoid, n from tanh (in-place on h)
__global__ void gru_combine(const float* __restrict__ gi,
                            const float* __restrict__ gh,
                            float* __restrict__ h)
{
  int idx = blockIdx.x * blockDim.x + threadIdx.x;
  if (idx >= kB * kHid) return;
  int b = idx >> 9;          // /512
  int c = idx & (kHid - 1);
  const float* gib = gi + (size_t)b * kG3;
  const float* ghb = gh + (size_t)b * kG3;
  float r = 1.f / (1.f + expf(-(gib[c] + ghb[c])));
  float z = 1.f / (1.f + expf(-(gib[kHid + c] + ghb[kHid + c])));
  float n = tanhf(gib[2 * kHid + c] + r * ghb[2 * kHid + c]);
  h[idx] = (1.f - z) * n + z * h[idx];
}

// x[b] = concat(src_emb[inp], trg_emb[wtok], bits)   (inp from state flags)
__global__ void build_x(const int* __restrict__ src,
                        const float* __restrict__ semb,
                        const float* __restrict__ temb,
                        const int* __restrict__ st,
                        const float* __restrict__ bits,
                        float* __restrict__ x)
{
  int idx = blockIdx.x * blockDim.x + threadIdx.x;
  if (idx >= kB * kInDim) return;
  int b = idx / kInDim;
  int c = idx - b * kInDim;
  float v;
  if (c < kEmb) {
    int tok = src[st[IS_I + b] * kB + b];
    if (st[IS_WRITING + b]) tok = SRC_NULLc;
    if (st[IS_NAUGHTY + b]) tok = SRC_EOSc;
    v = semb[(size_t)tok * kEmb + c];
  } else if (c < 2 * kEmb) {
    v = temb[(size_t)st[IS_WTOK + b] * kEmb + (c - kEmb)];
  } else {
    v = bits[b * 4 + (c - 2 * kEmb)];
  }
  x[idx] = v;
}

// Per-batch argmax over vlog (first-occurrence ties, as jnp.argmax) + logsumexp.
__global__ void reduce_vlog(const float* __restrict__ out,
                            int* __restrict__ st, float* __restrict__ logZ)
{
  const int b = blockIdx.x;
  const int tid = threadIdx.x;
  const float* row = out + (size_t)b * kNOut;
  __shared__ float smax[256];
  __shared__ int   sidx[256];
  __shared__ float ssum[256];

  float m = -INFINITY; int mi = 0;
  for (int c = tid; c < kVT; c += 256) {
    float v = row[c];
    if (v > m) { m = v; mi = c; }     // strict > keeps smallest index per thread
  }
  smax[tid] = m; sidx[tid] = mi; __syncthreads();
  for (int s = 128; s > 0; s >>= 1) {
    if (tid < s) {
      if (smax[tid + s] > smax[tid] ||
          (smax[tid + s] == smax[tid] && sidx[tid + s] < sidx[tid])) {
        smax[tid] = smax[tid + s]; sidx[tid] = sidx[tid + s];
      }
    }
    __syncthreads();
  }
  const float gm = smax[0];
  float acc = 0.f;
  for (int c = tid; c < kVT; c += 256) acc += expf(row[c] - gm);
  ssum[tid] = acc; __syncthreads();
  for (int s = 128; s > 0; s >>= 1) {
    if (tid < s) ssum[tid] += ssum[tid + s];
    __syncthreads();
  }
  if (tid == 0) { st[IS_WNEW + b] = sidx[0]; logZ[b] = gm + logf(ssum[0]); }
}

// Per-step carry update: one wave, lane = batch element.
__global__ void state_update(int t,
                             const int* __restrict__ src,
                             const int* __restrict__ trg,
                             const float* __restrict__ out,
                             const float* __restrict__ logZ,
                             int* __restrict__ st,
                             float* __restrict__ bits,
                             float* __restrict__ rew,
                             float* __restrict__ Qu)
{
  const int b = threadIdx.x;            // 0..31
  const float* row = out + (size_t)b * kNOut;
  float q0 = row[kVT], q1 = row[kVT + 1], q2 = row[kVT + 2];
  int action = 0; float qa = q0;
  if (q1 > qa) { qa = q1; action = 1; }
  if (q2 > qa) { qa = q2; action = 2; }

  const int term = st[IS_TERM + b];
  Qu[t * kB + b] = term ? 0.f : qa;     // Q_used (uses pre-update term)

  const int nt      = !term;
  const int reading = nt && (action == 0);
  const int writing = nt && (action == 1);
  const int bothing = nt && (action == 2);

  __shared__ int cnt[3];
  if (b < 3) cnt[b] = 0;
  __syncthreads();
  if (reading) atomicAdd(&cnt[0], 1);
  if (writing) atomicAdd(&cnt[1], 1);
  if (bothing) atomicAdd(&cnt[2], 1);
  __syncthreads();
  if (b < 3) st[IS_ACNT + b] += cnt[b];

  const int outp = writing || bothing;
  const int i = st[IS_I + b], j = st[IS_J + b];
  const int tgt = trg[j * kB + b];                     // == trg[old_j]
  const int just_term = outp && (tgt == TRG_EOSc);
  const int naughty = (reading || bothing) && (src[i * kB + b] == SRC_EOSc);
  int i_new = i + ((!naughty && (reading || bothing)) ? 1 : 0);
  if (i_new > kS - 1) i_new = kS - 1;
  int j_new = j + (outp ? 1 : 0);
  if (j_new > kT - 1) j_new = kT - 1;
  const int term_new = term || just_term;
  const int wtok = reading ? TRG_NULLc : st[IS_WNEW + b];

  bits[b * 4 + 0] = 0.f;
  bits[b * 4 + 1] = (src[i_new * kB + b] == SRC_EOSc) ? 1.f : 0.f;
  bits[b * 4 + 2] = outp ? 1.f : 0.f;
  bits[b * 4 + 3] = (reading || bothing) ? 1.f : 0.f;

  // reward = -CE(vlog, tgt) = vlog[tgt] - logZ (0 on PAD)
  rew[b] = (tgt == TRG_PADc) ? 0.f : (row[tgt] - logZ[b]);

  st[IS_I + b] = i_new;        st[IS_J + b] = j_new;
  st[IS_TERM + b] = term_new;  st[IS_WRITING + b] = writing;
  st[IS_NAUGHTY + b] = naughty; st[IS_WTOK + b] = wtok;
  st[IS_OUTP + b] = outp;      st[IS_OLDJ + b] = j;
  st[IS_READING + b] = reading; st[IS_JUSTTERM + b] = just_term;
}

// wouts[old_j, b, :] = vlog[b, :]  (only where outp)
__global__ void wouts_scatter(const float* __restrict__ out,
                              const int* __restrict__ st,
                              float* __restrict__ wouts)
{
  const int b = blockIdx.x;
  if (!st[IS_OUTP + b]) return;
  const int j = st[IS_OLDJ + b];
  float* dst = wouts + ((size_t)j * kB + b) * kVT;
  const float* srow = out + (size_t)b * kNOut;
  for (int c = threadIdx.x; c < kVT; c += blockDim.x) dst[c] = srow[c];
}

// Q_target from next step's q-columns (out holds out_{t+1} here).
__global__ void finalize_qt(int t,
                            const float* __restrict__ out,
                            const int* __restrict__ st,
                            const float* __restrict__ rew,
                            float* __restrict__ Qt)
{
  const int b = threadIdx.x;
  const float* row = out + (size_t)b * kNOut;
  const float nb = fmaxf(row[kVT], fmaxf(row[kVT + 1], row[kVT + 2]));
  const float reward = rew[b];
  float qt = reward + DISCOUNTc * nb;
  if (st[IS_TERM + b])      qt = 0.f;                  // updated term
  if (st[IS_READING + b])   qt = nb;
  if (st[IS_READING + b] && st[IS_NAUGHTY + b]) qt = DISCOUNTc * nb;
  if (st[IS_JUSTTERM + b])  qt = reward;
  qt -= Mc * (st[IS_NAUGHTY + b] ? 1.f : 0.f);
  Qt[t * kB + b] = qt;
}

__global__ void init_state(int* __restrict__ st, float* __restrict__ bits,
                           float* __restrict__ h0, float* __restrict__ h1)
{
  const int tid = threadIdx.x;
  for (int k = tid; k < kB * kHid; k += blockDim.x) { h0[k] = 0.f; h1[k] = 0.f; }
  if (tid < kB) {
    st[IS_I + tid] = 0;       st[IS_J + tid] = 0;
    st[IS_TERM + tid] = 0;    st[IS_WRITING + tid] = 0;
    st[IS_NAUGHTY + tid] = 0; st[IS_WTOK + tid] = TRG_NULLc;
    st[IS_OUTP + tid] = 0;    st[IS_OLDJ + tid] = 0;
    st[IS_READING + tid] = 0; st[IS_JUSTTERM + tid] = 0;
    st[IS_WNEW + tid] = TRG_NULLc;
    bits[tid * 4 + 0] = 1.f; bits[tid * 4 + 1] = 0.f;
    bits[tid * 4 + 2] = 0.f; bits[tid * 4 + 3] = 1.f;  // bits0 = [1,0,0,1]
  }
  if (tid < 3) st[IS_ACNT + tid] = 0;
}

__global__ void write_acnt(const int* __restrict__ st, float* __restrict__ dst)
{
  const int t = threadIdx.x;
  if (t < 3) dst[t] = (float)st[IS_ACNT + t];
}

// ---------------------------------------------------------------------------
extern "C" void kernel_launch(void* const* d_in, const int* in_sizes, int n_in,
                              void* d_out, int out_size, void* d_ws, size_t ws_size,
                              hipStream_t stream)
{
  (void)in_sizes; (void)n_in; (void)out_size; (void)ws_size;

  const int*   src  = (const int*)d_in[0];
  const int*   trg  = (const int*)d_in[1];
  const float* semb = (const float*)d_in[2];
  const float* temb = (const float*)d_in[3];
  const float* Wih0 = (const float*)d_in[4];
  const float* Whh0 = (const float*)d_in[5];
  const float* bih0 = (const float*)d_in[6];
  const float* bhh0 = (const float*)d_in[7];
  const float* Wih1 = (const float*)d_in[8];
  const float* Whh1 = (const float*)d_in[9];
  const float* bih1 = (const float*)d_in[10];
  const float* bhh1 = (const float*)d_in[11];
  const float* Wout = (const float*)d_in[12];
  const float* bout = (const float*)d_in[13];

  float* outp  = (float*)d_out;
  float* wouts = outp;                               // 48*32*16000
  float* Qu    = outp + (size_t)kT * kB * kVT;       // 95*32
  float* Qt    = Qu + (size_t)kSteps * kB;           // 95*32
  float* acnt  = Qt + (size_t)kSteps * kB;           // 3

  // workspace layout (floats): ~2.6 MB total
  float* ws   = (float*)d_ws;
  float* x    = ws;                                  // 32*516
  float* gi   = x    + (size_t)kB * kInDim;          // 32*1536
  float* gh   = gi   + (size_t)kB * kG3;             // 32*1536
  float* h0   = gh   + (size_t)kB * kG3;             // 32*512
  float* h1   = h0   + (size_t)kB * kHid;            // 32*512
  float* obuf = h1   + (size_t)kB * kHid;            // 32*16003
  float* logZ = obuf + (size_t)kB * kNOut;           // 32
  float* rew  = logZ + kB;                           // 32
  float* bits = rew  + kB;                           // 32*4
  int*   st   = (int*)(bits + kB * 4);               // IS_TOTAL ints

  hipMemsetAsync(wouts, 0, (size_t)kT * kB * kVT * sizeof(float), stream);
  init_state<<<1, 1024, 0, stream>>>(st, bits, h0, h1);

  auto net_step = [&]() {
    build_x<<<(kB * kInDim + 255) / 256, 256, 0, stream>>>(src, semb, temb, st, bits, x);
    gemm32_wmma<<<kG3 / 16, 32, 0, stream>>>(x,  Wih0, bih0, gi, kG3, kInDim);
    gemm32_wmma<<<kG3 / 16, 32, 0, stream>>>(h0, Whh0, bhh0, gh, kG3, kHid);
    gru_combine<<<(kB * kHid + 255) / 256, 256, 0, stream>>>(gi, gh, h0);
    gemm32_wmma<<<kG3 / 16, 32, 0, stream>>>(h0, Wih1, bih1, gi, kG3, kHid);
    gemm32_wmma<<<kG3 / 16, 32, 0, stream>>>(h1, Whh1, bhh1, gh, kG3, kHid);
    gru_combine<<<(kB * kHid + 255) / 256, 256, 0, stream>>>(gi, gh, h1);
    gemm32_wmma<<<(kNOut + 15) / 16, 32, 0, stream>>>(h1, Wout, bout, obuf, kNOut, kHid);
  };

  net_step();                                        // out_0 (h updated)
  for (int t = 0; t < kSteps; ++t) {
    reduce_vlog<<<kB, 256, 0, stream>>>(obuf, st, logZ);
    state_update<<<1, 32, 0, stream>>>(t, src, trg, obuf, logZ, st, bits, rew, Qu);
    wouts_scatter<<<kB, 256, 0, stream>>>(obuf, st, wouts);
    net_step();                                      // out_{t+1} == nout_t
    finalize_qt<<<1, 32, 0, stream>>>(t, obuf, st, rew, Qt);
  }
  write_acnt<<<1, 32, 0, stream>>>(st, acnt);
}